// AnisotropicDilatedProjectM2_18880676233726
// MI455X (gfx1250) — compile-verified
//
#include <hip/hip_runtime.h>
#include <math.h>

#define AS1 __attribute__((address_space(1)))
#define AS3 __attribute__((address_space(3)))

typedef int v2i __attribute__((__vector_size__(2 * sizeof(int))));

namespace {
constexpr int B = 8, C = 16, OR = 8, H = 256, W = 256;
constexpr int R = 5, K = 11;
constexpr int TH = 32, TW = 32;
constexpr int PH = TH + 2 * R;        // 42 rows per plane
constexpr int LPITCH = 44;            // words per row: covers gx = x0-6 .. x0+37
constexpr int CHUNKS = LPITCH / 2;    // 22 x B64 chunks per row (8B aligned)

// Correctly-rounded doubles of cos/sin at k*pi/8 (matches glibc/numpy; the only
// exact-boundary lattice taps occur at theta in {0, pi/2} where the reference's
// ~6.12e-17 residues round away in double).
constexpr double C8 = 0.92387953251128675613;   // cos(pi/8)
constexpr double S8 = 0.38268343236508977173;   // sin(pi/8)
constexpr double R2 = 0.70710678118654752440;   // sqrt(2)/2

struct Taps {
  bool  valid[OR][K][K];   // [o][ky][kx]
  float cost[OR][K][K];
};

constexpr Taps makeTaps() {
  Taps t{};
  const double cts[OR] = {1.0, C8, R2, S8, 0.0, -S8, -R2, -C8};
  const double sts[OR] = {0.0, S8, R2, C8, 1.0,  C8,  R2,  S8};
  constexpr double A = 2.0 / 3.0;
  constexpr double TWOA = 2.0 * A;
  constexpr double Q = TWOA - 1.0;     // exact (Sterbenz)
  constexpr double COEF = Q / TWOA;    // ~0.25; p/2 ~= 2 + 4.4e-16 -> rho2^2 faithful
  for (int o = 0; o < OR; ++o) {
    for (int ky = 0; ky < K; ++ky) {
      for (int kx = 0; kx < K; ++kx) {
        const double dy = (double)(ky - R), dx = (double)(kx - R);
        const double lon = cts[o] * dx + sts[o] * dy;
        const double lat = -sts[o] * dx + cts[o] * dy;
        const double rho2 = (lon / 5.0) * (lon / 5.0) + (lat / 2.5) * (lat / 2.5);
        t.valid[o][ky][kx] = (rho2 <= 1.0);
        t.cost[o][ky][kx] = (rho2 <= 1.0) ? (float)(COEF * (rho2 * rho2)) : 0.0f;
      }
    }
  }
  return t;
}
constexpr Taps TT = makeTaps();
}

// --- CDNA5 async global->LDS DMA (ASYNCcnt path) -------------------------
__device__ __forceinline__ void async_load_2xf32(const float* g, float* l) {
#if __has_builtin(__builtin_amdgcn_global_load_async_to_lds_b64)
  __builtin_amdgcn_global_load_async_to_lds_b64(
      (AS1 v2i*)g, (AS3 v2i*)l, 0, 0);
#else
  unsigned lds = (unsigned)(unsigned long long)l;
  unsigned long long ga = (unsigned long long)g;
  asm volatile("global_load_async_to_lds_b64 %0, %1, off"
               :: "v"(lds), "v"(ga) : "memory");
#endif
}

__device__ __forceinline__ void wait_async0() {
#if __has_builtin(__builtin_amdgcn_s_wait_asynccnt)
  __builtin_amdgcn_s_wait_asynccnt(0);
#else
  asm volatile("s_wait_asynccnt 0" ::: "memory");
#endif
}

// --- Main kernel: async-stage 8 orientation planes, max-plus reduce ------
__global__ __launch_bounds__(256)
void aniso_dilate_kernel(const float* __restrict__ x, float* __restrict__ out) {
  __shared__ __attribute__((aligned(16))) float tile[OR * PH * LPITCH]; // 59,136 B

  const int tid = threadIdx.x;
  const int x0 = blockIdx.x * TW;
  const int y0 = blockIdx.y * TH;
  const int bc = blockIdx.z;                       // fused (b, c)

  // stage 8 halo planes into LDS via async DMA, B64 chunks
  const float* xb = x + (size_t)bc * (size_t)(OR * H * W);
  const float NEG_INF = -__builtin_inff();
  for (int i = tid; i < OR * PH * CHUNKS; i += 256) {
    const int o = i / (PH * CHUNKS);
    const int rem = i % (PH * CHUNKS);
    const int r = rem / CHUNKS;
    const int ch = rem % CHUNKS;
    const int gy = y0 - R + r;
    const int gx0 = x0 - 6 + ch * 2;               // even -> 8B aligned, all-in/all-out
    float* lp = &tile[(o * PH + r) * LPITCH + ch * 2];
    if ((unsigned)gy < (unsigned)H && (unsigned)gx0 < (unsigned)W) {
      async_load_2xf32(xb + (size_t)o * (H * W) + (size_t)gy * W + gx0, lp);
    } else {
      lp[0] = NEG_INF;
      lp[1] = NEG_INF;
    }
  }
  wait_async0();          // drain ASYNCcnt (our own LDS writes)
  __syncthreads();        // cross-wave visibility (s_barrier_signal/wait)

  // max-plus reduction: 4 consecutive rows per thread (col tx, rows ty0..ty0+3)
  // Fully unrolled with compile-time taps: constant DS offsets, CSE'd row loads.
  const int tx = tid & 31;
  const int ty0 = (tid >> 5) * 4;
  const float* tb = &tile[ty0 * LPITCH + tx];
  float m0 = NEG_INF, m1 = NEG_INF, m2 = NEG_INF, m3 = NEG_INF;
#pragma unroll
  for (int o = 0; o < OR; ++o) {
#pragma unroll
    for (int kx = 0; kx < K; ++kx) {
#pragma unroll
      for (int ky = 0; ky < K; ++ky) {
        if (TT.valid[o][ky][kx]) {
          const float c = TT.cost[o][ky][kx];
          const int a = o * PH * LPITCH + ky * LPITCH + (kx + 1);
          m0 = fmaxf(m0, tb[a] - c);
          m1 = fmaxf(m1, tb[a + LPITCH] - c);
          m2 = fmaxf(m2, tb[a + 2 * LPITCH] - c);
          m3 = fmaxf(m3, tb[a + 3 * LPITCH] - c);
        }
      }
    }
  }

  const size_t ob = (size_t)bc * (size_t)(H * W) + (size_t)(y0 + ty0) * W + (x0 + tx);
  out[ob]         = m0;
  out[ob + W]     = m1;
  out[ob + 2 * W] = m2;
  out[ob + 3 * W] = m3;
}

extern "C" void kernel_launch(void* const* d_in, const int* in_sizes, int n_in,
                              void* d_out, int out_size, void* d_ws, size_t ws_size,
                              hipStream_t stream) {
  (void)in_sizes; (void)n_in; (void)out_size; (void)d_ws; (void)ws_size;
  const float* x = (const float*)d_in[0];
  float* out = (float*)d_out;

  dim3 grid(W / TW, H / TH, B * C);
  aniso_dilate_kernel<<<grid, dim3(256), 0, stream>>>(x, out);
}